// CovidModel_52037823758535
// MI455X (gfx1250) — compile-verified
//
#include <hip/hip_runtime.h>

#define B_SERIES 16384
#define F_DAYS   512
#define W_WARM   14
#define J_WIN    10
#define LSTRIDE  524                 // J + F + 2 pad (max LDS t index accessed = 523)
#define WAVES_PER_BLOCK 2
#define SERIES_PER_WAVE 16

typedef __attribute__((ext_vector_type(2))) float v2f;
typedef __attribute__((ext_vector_type(8))) float v8f;

// Per-lane transition weight: lane i holds w_i (Poisson-PMF weight for delay i),
// zero outside 1..10. Later gathered with __shfl (ds_bpermute) to build banded B tiles.
__device__ __forceinline__ float lane_weight(float u_rho, float u_lam, float u_nu, int lane) {
    float rho = 1.0f / (1.0f + __expf(-u_rho));          // sigmoid
    float nu  = __logf(1.0f + __expf(u_nu));             // softplus
    float lam = __logf(1.0f + __expf(u_lam));            // softplus
    float j   = (float)lane;
    float pmf = __expf(j * __logf(lam) - lam - lgammaf(j + 1.0f));
    return (lane >= 1 && lane <= J_WIN) ? (rho * nu * pmf) : 0.0f;
}

// One causal FIR conv over 512 days for 16 series via V_WMMA_F32_16X16X4_F32.
// rd: LDS buffer holding input series at t = day + J (t<J = warmup history, pads zero).
// If gout == nullptr, write result into wr (same t layout); else write to global out.
__device__ __forceinline__ void do_conv(const float* rd, float* wr, float* gout,
                                        float wlane, int lane, int sbase_global) {
    const int n  = lane & 15;   // output day within tile / B-matrix column / A series row
    const int kh = lane >> 4;   // lane-half selects K pair {0,1} vs {2,3}

    // Banded weight tiles: B_tt[k][n] = w(10 - 4*tt + n - k), 7 K-tiles cover days d0-10..d0+17.
    v2f bw[7];
#pragma unroll
    for (int tt = 0; tt < 7; ++tt) {
        int j0 = 10 - 4 * tt + n - 2 * kh;   // K = 2*kh   (VGPR0 row)
        int j1 = j0 - 1;                     // K = 2*kh+1 (VGPR1 row)
        int c0 = j0 < 0 ? 0 : j0;            // max 25 < 32; lane 0 holds 0.0f
        int c1 = j1 < 0 ? 0 : j1;
        bw[tt].x = __shfl(wlane, c0, 32);
        bw[tt].y = __shfl(wlane, c1, 32);
    }

    for (int d0 = 0; d0 < F_DAYS; d0 += 16) {
        v8f acc = {0.f, 0.f, 0.f, 0.f, 0.f, 0.f, 0.f, 0.f};
#pragma unroll
        for (int tt = 0; tt < 7; ++tt) {
            // A 16x4 f32 layout: lane%16 = series (M), VGPR0/1 = K {2*kh, 2*kh+1}
            // LDS t index of K-tile start: (d0 - 10 + 4*tt) + J = d0 + 4*tt
            const v2f av = *(const v2f*)(rd + (size_t)n * LSTRIDE + (d0 + 4 * tt + 2 * kh));
            acc = __builtin_amdgcn_wmma_f32_16x16x4_f32(
                false, av, false, bw[tt], (short)0, acc, false, false);
        }
        if (gout) {
#pragma unroll
            for (int r = 0; r < 8; ++r)      // D: M = r + 8*kh, day = d0 + n
                gout[(size_t)(sbase_global + r + 8 * kh) * F_DAYS + d0 + n] = acc[r];
        } else {
#pragma unroll
            for (int r = 0; r < 8; ++r)
                wr[(size_t)(r + 8 * kh) * LSTRIDE + (J_WIN + d0 + n)] = acc[r];
        }
    }
}

__global__ __launch_bounds__(WAVES_PER_BLOCK * 32) void covid_admissions_kernel(
    const float* __restrict__ rt,
    const float* __restrict__ wua,  // warmup_asymp   (B, 14)
    const float* __restrict__ wum,  // warmup_mild    (B, 14)
    const float* __restrict__ wux,  // warmup_extreme (B, 14)
    const float* __restrict__ urM, const float* __restrict__ ulM, const float* __restrict__ unM,
    const float* __restrict__ urX, const float* __restrict__ ulX, const float* __restrict__ unX,
    const float* __restrict__ urG, const float* __restrict__ ulG, const float* __restrict__ unG,
    float* __restrict__ out)
{
    __shared__ float lds_a[WAVES_PER_BLOCK][SERIES_PER_WAVE * LSTRIDE];
    __shared__ float lds_m[WAVES_PER_BLOCK][SERIES_PER_WAVE * LSTRIDE];

    const int tid   = threadIdx.x;
    const int wv    = tid >> 5;
    const int lane  = tid & 31;
    const int sbase = (blockIdx.x * WAVES_PER_BLOCK + wv) * SERIES_PER_WAVE;
    float* A = lds_a[wv];
    float* M = lds_m[wv];
    const float invT = 1.0f / 5.8f;

    // ---- phase 0: s[d] = rt^(1/T) into M (temp, t=d); asymp history into A; zero pads.
    for (int i = 0; i < (SERIES_PER_WAVE * F_DAYS) / 32; ++i) {
        int idx = i * 32 + lane;
        int s = idx >> 9;       // /512
        int d = idx & 511;
        float r = rt[(size_t)(sbase + s) * F_DAYS + d];
        M[s * LSTRIDE + d] = __expf(invT * __logf(r));
    }
    for (int idx = lane; idx < SERIES_PER_WAVE * J_WIN; idx += 32) {
        int s = idx / J_WIN, t = idx % J_WIN;   // a[t-J] = warmup[W-J+t]
        A[s * LSTRIDE + t] = wua[(size_t)(sbase + s) * W_WARM + (W_WARM - J_WIN) + t];
    }
    {   // zero the 2 pad slots (t=522,523) of both buffers for every series
        int s = lane & 15;
        int t = (J_WIN + F_DAYS) + (lane >> 4);
        A[s * LSTRIDE + t] = 0.0f;
        M[s * LSTRIDE + t] = 0.0f;
    }
    __syncthreads();

    // ---- phase 1: serial multiplicative scan a[d] = a[d-1]*s[d], lanes 0..15.
    if (lane < 16) {
        int s = lane;
        float a = wua[(size_t)(sbase + s) * W_WARM + (W_WARM - 1)];   // a[-1]
        const float* Ms = M + s * LSTRIDE;
        float*       As = A + s * LSTRIDE;
        for (int d = 0; d < F_DAYS; ++d) {
            a *= Ms[d];
            As[J_WIN + d] = a;
        }
    }
    __syncthreads();

    // ---- phase 2: mild history into M (overwrites spent s-values), conv a -> m.
    for (int idx = lane; idx < SERIES_PER_WAVE * J_WIN; idx += 32) {
        int s = idx / J_WIN, t = idx % J_WIN;
        M[s * LSTRIDE + t] = wum[(size_t)(sbase + s) * W_WARM + (W_WARM - J_WIN) + t];
    }
    __syncthreads();
    float wl = lane_weight(urM[0], ulM[0], unM[0], lane);
    do_conv(A, M, nullptr, wl, lane, sbase);
    __syncthreads();

    // ---- phase 3: extreme history into A (a no longer needed), conv m -> x.
    for (int idx = lane; idx < SERIES_PER_WAVE * J_WIN; idx += 32) {
        int s = idx / J_WIN, t = idx % J_WIN;
        A[s * LSTRIDE + t] = wux[(size_t)(sbase + s) * W_WARM + (W_WARM - J_WIN) + t];
    }
    __syncthreads();
    wl = lane_weight(urX[0], ulX[0], unX[0], lane);
    do_conv(M, A, nullptr, wl, lane, sbase);
    __syncthreads();

    // ---- phase 4: conv x -> g, stream straight to global output.
    wl = lane_weight(urG[0], ulG[0], unG[0], lane);
    do_conv(A, nullptr, out, wl, lane, sbase);
}

extern "C" void kernel_launch(void* const* d_in, const int* in_sizes, int n_in,
                              void* d_out, int out_size, void* d_ws, size_t ws_size,
                              hipStream_t stream) {
    (void)in_sizes; (void)n_in; (void)d_ws; (void)ws_size; (void)out_size;
    const float* rt  = (const float*)d_in[0];
    const float* wua = (const float*)d_in[1];
    const float* wum = (const float*)d_in[2];
    const float* wux = (const float*)d_in[3];
    const float* urM = (const float*)d_in[4];
    const float* ulM = (const float*)d_in[5];
    const float* unM = (const float*)d_in[6];
    const float* urX = (const float*)d_in[7];
    const float* ulX = (const float*)d_in[8];
    const float* unX = (const float*)d_in[9];
    const float* urG = (const float*)d_in[10];
    const float* ulG = (const float*)d_in[11];
    const float* unG = (const float*)d_in[12];
    float* out = (float*)d_out;

    const int blocks = B_SERIES / (WAVES_PER_BLOCK * SERIES_PER_WAVE);   // 512
    covid_admissions_kernel<<<blocks, WAVES_PER_BLOCK * 32, 0, stream>>>(
        rt, wua, wum, wux, urM, ulM, unM, urX, ulX, unX, urG, ulG, unG, out);
}